// HTMModel_19834158973432
// MI455X (gfx1250) — compile-verified
//
#include <hip/hip_runtime.h>

typedef float v2f __attribute__((ext_vector_type(2)));
typedef float v4f __attribute__((ext_vector_type(4)));
typedef float v8f __attribute__((ext_vector_type(8)));

constexpr int NUM_COLS = 4096;     // minicolumns (rows of connections)
constexpr int IN       = 32768;    // input size (K)
constexpr int NSEG     = 32;       // K segments per row-tile
constexpr int SEG      = IN / NSEG;        // 1024
constexpr int TILES    = NUM_COLS / 16;    // 256
constexpr int WAVES_PER_BLOCK = 8;         // 256 threads / wave32
constexpr int NUM_ACTIVE = 82;

// ---------------------------------------------------------------------------
// Zero the overlap accumulator in workspace (graph-capture safe, no memset).
// ---------------------------------------------------------------------------
__global__ void zero_kernel(float* __restrict__ p, int n) {
    int i = blockIdx.x * blockDim.x + threadIdx.x;
    if (i < n) p[i] = 0.0f;
}

// ---------------------------------------------------------------------------
// overlap[c] = sum_i conn[c,i] * x[i]   via V_WMMA_F32_16X16X4_F32.
// One wave handles a 16-row tile over one 1024-wide K segment.
//   A slot (lane m = lane&15, half h = lane>>4): conn[row+m][kb + 4h + {0,1}]
//   and the second WMMA of the pair takes {2,3} — B uses the identical per-slot
//   K assignment, so the hardware's internal K ordering cancels out.
//   B is replicated across all 16 columns -> every column of C equals the
//   per-row partial sum; lanes with m==0 (lane 0 / lane 16) flush 8 rows each.
// conn is a 512 MB single-use stream (> 192 MB L2): load it non-temporal so
// the hot 128 KB x vector and the accumulator stay L2-resident.
// ---------------------------------------------------------------------------
__global__ __launch_bounds__(256)
void overlap_wmma_kernel(const float* __restrict__ conn,
                         const float* __restrict__ x,
                         float* __restrict__ ovl) {
    const int lane = threadIdx.x & 31;
    const int wid  = blockIdx.x * WAVES_PER_BLOCK + (threadIdx.x >> 5);
    const int tile = wid / NSEG;          // 0..255
    const int seg  = wid % NSEG;          // 0..31
    const int m    = lane & 15;
    const int h    = lane >> 4;           // 0 or 1 (wave half)

    const float* __restrict__ arow = conn + (size_t)(tile * 16 + m) * IN + seg * SEG;
    const float* __restrict__ xseg = x + seg * SEG;

    v8f c = {};
#pragma unroll 4
    for (int kb = 0; kb < SEG; kb += 8) {
        const v4f av = __builtin_nontemporal_load(
            reinterpret_cast<const v4f*>(arow + kb + 4 * h));        // streamed
        const v4f bv = *reinterpret_cast<const v4f*>(xseg + kb + 4 * h); // cached
        v2f a0 = { av.x, av.y };
        v2f a1 = { av.z, av.w };
        v2f b0 = { bv.x, bv.y };
        v2f b1 = { bv.z, bv.w };
        // (neg_a, A, neg_b, B, c_mod, C, reuse_a, reuse_b)
        c = __builtin_amdgcn_wmma_f32_16x16x4_f32(false, a0, false, b0,
                                                  (short)0, c, false, false);
        c = __builtin_amdgcn_wmma_f32_16x16x4_f32(false, a1, false, b1,
                                                  (short)0, c, false, false);
    }

    // C layout: VGPR j -> M=j (lanes 0-15) / M=j+8 (lanes 16-31); all N equal.
    if (m == 0) {
        const int rbase = tile * 16 + 8 * h;
#pragma unroll
        for (int j = 0; j < 8; ++j)
            atomicAdd(&ovl[rbase + j], c[j]);
    }
}

// ---------------------------------------------------------------------------
// Exact top-82 with lower-index tie-breaking (matches jax.lax.top_k), then
// emit active mask and masked boosted overlap. Values are >= 0, so fp32 bit
// patterns compare as unsigned ints. 4-pass radix select + block prefix scan.
// ---------------------------------------------------------------------------
__global__ __launch_bounds__(1024)
void select_kernel(const float* __restrict__ ovl,
                   const float* __restrict__ boost,
                   float* __restrict__ out) {
    constexpr int TPB = 1024;
    constexpr int PER = NUM_COLS / TPB;   // 4

    __shared__ unsigned int hist[256];
    __shared__ unsigned int scan[TPB];
    __shared__ unsigned int s_prefix;
    __shared__ unsigned int s_want;
    __shared__ unsigned int s_gt;

    const int t = threadIdx.x;

    float        bval[PER];
    unsigned int key[PER];
#pragma unroll
    for (int j = 0; j < PER; ++j) {
        const int i = t * PER + j;
        const float v = ovl[i] * boost[i];
        bval[j] = v;
        key[j]  = __float_as_uint(v);     // v >= 0: uint order == float order
    }

    if (t == 0) { s_prefix = 0u; s_want = NUM_ACTIVE; s_gt = 0u; }
    __syncthreads();

    // Radix select: find exact key of the 82nd largest element.
    for (int pass = 0; pass < 4; ++pass) {
        const int shift = 24 - 8 * pass;
        if (t < 256) hist[t] = 0u;
        __syncthreads();
        const unsigned int prefix = s_prefix;
        const unsigned int maskHi = (pass == 0) ? 0u : (0xFFFFFFFFu << (shift + 8));
#pragma unroll
        for (int j = 0; j < PER; ++j)
            if ((key[j] & maskHi) == (prefix & maskHi))
                atomicAdd(&hist[(key[j] >> shift) & 0xFFu], 1u);
        __syncthreads();
        if (t == 0) {
            unsigned int want = s_want, cum = 0u;
            int b = 255;
            for (; b > 0; --b) {
                const unsigned int hcnt = hist[b];
                if (cum + hcnt >= want) break;
                cum += hcnt;
            }
            s_want   = want - cum;
            s_prefix = prefix | ((unsigned int)b << shift);
        }
        __syncthreads();
    }
    const unsigned int T = s_prefix;

    // Count strictly-greater and per-thread equal counts.
    unsigned int localGt = 0u, localEq = 0u;
#pragma unroll
    for (int j = 0; j < PER; ++j) {
        if (key[j] > T)       ++localGt;
        else if (key[j] == T) ++localEq;
    }
    atomicAdd(&s_gt, localGt);
    scan[t] = localEq;
    __syncthreads();

    // Block-wide inclusive scan (Hillis-Steele, two-barrier).
    for (int off = 1; off < TPB; off <<= 1) {
        const unsigned int mine  = scan[t];
        const unsigned int other = (t >= off) ? scan[t - off] : 0u;
        __syncthreads();
        scan[t] = mine + other;
        __syncthreads();
    }

    const unsigned int rem = NUM_ACTIVE - s_gt;        // ties to accept
    unsigned int eqBefore  = scan[t] - localEq;         // exclusive prefix

#pragma unroll
    for (int j = 0; j < PER; ++j) {
        const int i = t * PER + j;
        bool act;
        if (key[j] > T) {
            act = true;
        } else if (key[j] == T) {
            act = (eqBefore < rem);
            ++eqBefore;
        } else {
            act = false;
        }
        out[i]            = act ? 1.0f    : 0.0f;
        out[NUM_COLS + i] = act ? bval[j] : 0.0f;
    }
}

// ---------------------------------------------------------------------------
extern "C" void kernel_launch(void* const* d_in, const int* in_sizes, int n_in,
                              void* d_out, int out_size, void* d_ws, size_t ws_size,
                              hipStream_t stream) {
    const float* x     = (const float*)d_in[0];   // input_vector [32768]
    const float* conn  = (const float*)d_in[1];   // connections  [4096, 32768]
    const float* boost = (const float*)d_in[2];   // boosting_factors [4096]
    float* out = (float*)d_out;                   // [8192] = mask ++ masked boosted
    float* ovl = (float*)d_ws;                    // [4096] overlap accumulator

    zero_kernel<<<(NUM_COLS + 255) / 256, 256, 0, stream>>>(ovl, NUM_COLS);

    const int nblocks = TILES * NSEG / WAVES_PER_BLOCK;   // 1024 blocks, 8192 waves
    overlap_wmma_kernel<<<nblocks, 256, 0, stream>>>(conn, x, ovl);

    select_kernel<<<1, 1024, 0, stream>>>(ovl, boost, out);
}